// Basenet_collective_15839839387966
// MI455X (gfx1250) — compile-verified
//
#include <hip/hip_runtime.h>
#include <hip/hip_bf16.h>
#include <math.h>
#include <stdint.h>

// ---------------------------------------------------------------------------
// Problem constants (from reference)
// ---------------------------------------------------------------------------
#define BT    80
#define MAXN  13
#define M_DIM (BT * MAXN)   // 1040 rows, divisible by 16
#define K2D   26400
#define NFB   1024
#define NA    6
#define NG    5

// GEMM tiling
#define BM 128
#define BN 128
#define BK 16
#define APAD 20    // LDS row stride (floats) for A tile: 20r mod 64 distinct -> no bank conflicts
#define BPAD 136   // LDS row stride (floats) for B tile: 2*136 mod 64 = 16 -> half-waves hit disjoint banks

typedef __attribute__((ext_vector_type(2))) float        v2f;
typedef __attribute__((ext_vector_type(8))) float        v8f;
typedef __attribute__((ext_vector_type(4))) int          v4i;
typedef __attribute__((ext_vector_type(8))) int          v8i;
typedef __attribute__((ext_vector_type(4))) unsigned int v4u;

// ---------------------------------------------------------------------------
// CDNA5 async global->LDS copy (ASYNCcnt) + TDM tensor load (TENSORcnt)
// ---------------------------------------------------------------------------
#if __has_builtin(__builtin_amdgcn_global_load_async_to_lds_b128)
#define HAVE_ASYNC_BUILTIN 1
#endif
#if __has_builtin(__builtin_amdgcn_tensor_load_to_lds)
#define HAVE_TDM 1
#endif

static __device__ __forceinline__ void async_cp16(const float* gsrc, float* ldst) {
#if defined(HAVE_ASYNC_BUILTIN)
    __attribute__((address_space(1))) v4i* g =
        (__attribute__((address_space(1))) v4i*)(unsigned long long)(uintptr_t)gsrc;
    __attribute__((address_space(3))) v4i* l =
        (__attribute__((address_space(3))) v4i*)(unsigned)(uintptr_t)ldst;
    __builtin_amdgcn_global_load_async_to_lds_b128(g, l, 0, 0);
#else
    unsigned           lo = (unsigned)(uintptr_t)ldst;            // LDS offset = addr[31:0]
    unsigned long long ga = (unsigned long long)(uintptr_t)gsrc;  // global address
    asm volatile("global_load_async_to_lds_b128 %0, %1, off" :: "v"(lo), "v"(ga) : "memory");
#endif
}

static __device__ __forceinline__ void wait_async0() {
#if __has_builtin(__builtin_amdgcn_s_wait_asynccnt)
    __builtin_amdgcn_s_wait_asynccnt(0);
#else
    asm volatile("s_wait_asynccnt 0x0" ::: "memory");
#endif
}

#if defined(HAVE_TDM)
// TDM: load one 16(K) x 128(N) f32 tile of W[26400,1024] into LDS laid out as
// Bs[16][136] (128-DWORD rows + 8-DWORD pad == pad_interval=6, pad_amount=7).
static __device__ __forceinline__ void tdm_load_B(const float* gsrc, float* ldst) {
    unsigned           lds = (unsigned)(uintptr_t)ldst;
    unsigned long long ga  = (unsigned long long)(uintptr_t)gsrc;
    // D# group 0: count=1 | lds_addr | global_addr[56:0] | type=2
    v4u g0;
    g0.x = 1u;                                          // count=1 (valid user descriptor)
    g0.y = lds;                                         // lds_addr (bytes)
    g0.z = (unsigned)(ga & 0xFFFFFFFFu);                // global_addr[31:0]
    g0.w = (unsigned)((ga >> 32) & 0x01FFFFFFu) | 0x80000000u; // addr[56:32] | type=2<<30
    // D# group 1 (bit-packed per ISA 8.4)
    v8i g1;
    g1[0] = (int)0x0F920000;     // pad_amount=7(->8DW), pad_interval=6(->128DW), pad_en=1, data_size=2(4B)
    g1[1] = (int)((NFB & 0xFFFF) << 16);                          // tensor_dim0[15:0] @bits63:48
    g1[2] = (int)(((NFB >> 16) & 0xFFFF) | ((K2D & 0xFFFF) << 16)); // dim0 hi | tensor_dim1 lo
    g1[3] = (int)(((K2D >> 16) & 0xFFFF) | (BN << 16));           // dim1 hi | tile_dim0=128
    g1[4] = BK;                                                   // tile_dim1=16, tile_dim2=0
    g1[5] = NFB;                                                  // tensor_dim0_stride = 1024
    g1[6] = 0;
    g1[7] = 0;
    v4i zz4 = (v4i)0;                                             // groups 2/3 unused (2-D tensor)
    v8i zz8 = (v8i)0;                                             // extra group (clang-23 6-arg form)
    __builtin_amdgcn_tensor_load_to_lds(g0, g1, zz4, zz4, zz8, 0);
}
#endif

static __device__ __forceinline__ void wait_tensor0() {
#if defined(HAVE_TDM)
    __builtin_amdgcn_s_wait_tensorcnt(0);
#endif
}

// ---------------------------------------------------------------------------
// Kernel 1: H = ReLU(A[1040,26400] @ W[26400,1024] + b_emb)   (FP32 WMMA)
//   grid = (9, 8), block = 256 threads = 8 waves; wave = 32x64 output tile.
//   Double-buffered pipeline: A tile via per-thread async-to-LDS (ASYNCcnt),
//   B tile via one TDM descriptor from wave 0 (TENSORcnt); overlap with WMMAs.
// ---------------------------------------------------------------------------
__global__ __launch_bounds__(256)
void gemm_emb_relu(const float* __restrict__ A,
                   const float* __restrict__ W,
                   const float* __restrict__ bias,
                   float* __restrict__ H) {
    __shared__ float As[2][BM][APAD];
    __shared__ float Bs[2][BK][BPAD];

    const int tid  = threadIdx.x;
    const int lane = tid & 31;
    const int wid  = tid >> 5;          // 0..7
    const int half = lane >> 4;         // 0/1 (lane group)
    const int r    = lane & 15;

    const int m0 = blockIdx.x * BM;
    const int n0 = blockIdx.y * BN;
    const int wm = (wid >> 1) * 32;     // wave M offset (0,32,64,96)
    const int wn = (wid & 1) * 64;      // wave N offset (0,64)

    // A-tile chunks this thread fills (2 x b128)
    int a_row[2], a_kq[2], a_grow[2];
#if !defined(HAVE_TDM)
    int b_row[2], b_nq[2];
#endif
    #pragma unroll
    for (int i = 0; i < 2; ++i) {
        int idx  = tid + i * 256;
        a_row[i] = idx >> 2;
        a_kq[i]  = idx & 3;
        int gr   = m0 + a_row[i];
        a_grow[i] = (gr > M_DIM - 1) ? (M_DIM - 1) : gr;  // clamp OOB rows (stores guarded)
#if !defined(HAVE_TDM)
        b_row[i] = idx >> 5;
        b_nq[i]  = idx & 31;
#endif
    }

    v8f acc[2][4];
    #pragma unroll
    for (int i = 0; i < 2; ++i)
        #pragma unroll
        for (int j = 0; j < 4; ++j)
            acc[i][j] = (v8f)0.0f;

    // ---- fill of one K-tile into buffer `buf` ----
    auto load_tile = [&](int k0, int buf) {
        #pragma unroll
        for (int i = 0; i < 2; ++i)
            async_cp16(A + (size_t)a_grow[i] * K2D + (k0 + a_kq[i] * 4),
                       &As[buf][a_row[i]][a_kq[i] * 4]);
#if defined(HAVE_TDM)
        if (wid == 0)
            tdm_load_B(W + (size_t)k0 * NFB + n0, &Bs[buf][0][0]);
#else
        #pragma unroll
        for (int i = 0; i < 2; ++i)
            async_cp16(W + (size_t)(k0 + b_row[i]) * NFB + (n0 + b_nq[i] * 4),
                       &Bs[buf][b_row[i]][b_nq[i] * 4]);
#endif
    };

    // ---- 4 k-steps of V_WMMA_F32_16X16X4_F32 on buffer `buf` ----
    auto compute_tile = [&](int buf) {
        #pragma unroll
        for (int kk = 0; kk < BK; kk += 4) {
            v2f a[2], b[4];
            #pragma unroll
            for (int mt = 0; mt < 2; ++mt) {
                a[mt].x = As[buf][wm + 16 * mt + r][kk + 2 * half + 0];
                a[mt].y = As[buf][wm + 16 * mt + r][kk + 2 * half + 1];
            }
            #pragma unroll
            for (int nt = 0; nt < 4; ++nt) {
                b[nt].x = Bs[buf][kk + 2 * half + 0][wn + 16 * nt + r];
                b[nt].y = Bs[buf][kk + 2 * half + 1][wn + 16 * nt + r];
            }
            #pragma unroll
            for (int mt = 0; mt < 2; ++mt)
                #pragma unroll
                for (int nt = 0; nt < 4; ++nt)
                    acc[mt][nt] = __builtin_amdgcn_wmma_f32_16x16x4_f32(
                        false, a[mt], false, b[nt], (short)0, acc[mt][nt], false, false);
        }
    };

    // ---- software pipeline: prefetch(i+1) || compute(i) ----
    int buf = 0;
    load_tile(0, 0);
    wait_tensor0();
    wait_async0();
    __syncthreads();
    for (int k0 = BK; k0 < K2D; k0 += BK) {
        load_tile(k0, buf ^ 1);   // async + TDM: overlap with WMMAs below
        compute_tile(buf);
        wait_tensor0();           // wave0's TDM fill of buf^1 complete
        wait_async0();            // own async fills of buf^1 complete
        __syncthreads();
        buf ^= 1;
    }
    compute_tile(buf);

    // ---- epilogue: + bias, ReLU, store (C/D layout: VGPR v -> M = v + 8*half, N = r) ----
    #pragma unroll
    for (int mt = 0; mt < 2; ++mt) {
        #pragma unroll
        for (int nt = 0; nt < 4; ++nt) {
            const int gcol = n0 + wn + 16 * nt + r;
            const float bv = bias[gcol];
            #pragma unroll
            for (int v = 0; v < 8; ++v) {
                const int grow = m0 + wm + 16 * mt + v + 8 * half;
                if (grow < M_DIM) {
                    float x = acc[mt][nt][v] + bv;
                    H[(size_t)grow * NFB + gcol] = fmaxf(x, 0.0f);
                }
            }
        }
    }
}

// ---------------------------------------------------------------------------
// Kernel 2: per-frame epilogue (one block per frame, 256 threads = 8 waves)
// ---------------------------------------------------------------------------
__global__ __launch_bounds__(256)
void heads_epilogue(const float* __restrict__ H,
                    const float* __restrict__ w_act,  const float* __restrict__ b_act,
                    const float* __restrict__ w_acty, const float* __restrict__ b_acty,
                    const int* __restrict__ bboxes_num,
                    float* __restrict__ actions,      // [BT, MAXN, NA]
                    float* __restrict__ activities) { // [BT, NG]
    __shared__ float pooled_s[NFB];

    const int bt   = blockIdx.x;
    const int nv   = bboxes_num[bt];
    const int wid  = threadIdx.x >> 5;
    const int lane = threadIdx.x & 31;

    // masked max-pool over valid boxes (nv >= 1 always)
    for (int f = threadIdx.x; f < NFB; f += 256) {
        float m = -3.402823466e38f; // finfo(f32).min
        for (int n = 0; n < nv; ++n)
            m = fmaxf(m, H[((size_t)(bt * MAXN + n)) * NFB + f]);
        pooled_s[f] = m;
    }
    __syncthreads();

    // action scores: 78 dot products of length 1024, one per wave-slot
    for (int p = wid; p < MAXN * NA; p += 8) {
        const int n = p / NA, a = p % NA;
        const float* hrow = H + ((size_t)(bt * MAXN + n)) * NFB;
        float s = 0.0f;
        for (int f = lane; f < NFB; f += 32)
            s += hrow[f] * w_act[f * NA + a];
        #pragma unroll
        for (int off = 16; off > 0; off >>= 1)
            s += __shfl_xor(s, off, 32);
        if (lane == 0)
            actions[(size_t)(bt * MAXN + n) * NA + a] = (n < nv) ? (s + b_act[a]) : 0.0f;
    }

    // activity scores: 5 dot products of pooled features
    for (int g = wid; g < NG; g += 8) {
        float s = 0.0f;
        for (int f = lane; f < NFB; f += 32)
            s += pooled_s[f] * w_acty[f * NG + g];
        #pragma unroll
        for (int off = 16; off > 0; off >>= 1)
            s += __shfl_xor(s, off, 32);
        if (lane == 0)
            activities[bt * NG + g] = s + b_acty[g];
    }
}

// ---------------------------------------------------------------------------
// Launch
// ---------------------------------------------------------------------------
extern "C" void kernel_launch(void* const* d_in, const int* in_sizes, int n_in,
                              void* d_out, int out_size, void* d_ws, size_t ws_size,
                              hipStream_t stream) {
    const float* A      = (const float*)d_in[0]; // [80,13,26400]
    const float* w_emb  = (const float*)d_in[1]; // [26400,1024]
    const float* b_emb  = (const float*)d_in[2]; // [1024]
    const float* w_act  = (const float*)d_in[3]; // [1024,6]
    const float* b_act  = (const float*)d_in[4]; // [6]
    const float* w_acty = (const float*)d_in[5]; // [1024,5]
    const float* b_acty = (const float*)d_in[6]; // [5]
    const int*   bnum   = (const int*)d_in[7];   // [80]

    float* H          = (float*)d_ws;            // [1040,1024] = 4.06 MB scratch
    float* actions    = (float*)d_out;           // [80,13,6] = 6240 floats
    float* activities = actions + BT * MAXN * NA;// [80,5]    =  400 floats

    dim3 g1((M_DIM + BM - 1) / BM, NFB / BN);    // (9, 8)
    gemm_emb_relu<<<g1, 256, 0, stream>>>(A, w_emb, b_emb, H);
    heads_epilogue<<<BT, 256, 0, stream>>>(H, w_act, b_act, w_acty, b_acty, bnum,
                                           actions, activities);
}